// FurthestPoint_48971217109122
// MI455X (gfx1250) — compile-verified
//
#include <hip/hip_runtime.h>
#include <cstdint>

// Furthest Point Sampling for MI455X (gfx1250, wave32).
// One 1024-thread workgroup per batch. dist[] (256KB) + pick buffers (32KB)
// live in CDNA5's 320KB-per-WGP LDS (register-resident dist spilled to
// scratch under the launch_bounds budget -- LDS is the right home for it).
// Wave32 shuffle argmax + 32-wave LDS tree; final drain via
// global_store_async_from_lds_b32 (ASYNCcnt path).

#define FPS_N        65536
#define FPS_NPOINT   2048
#define FPS_THREADS  1024
#define FPS_PPT      (FPS_N / FPS_THREADS)   // 64 points per thread

struct F3 { float x, y, z; };

__device__ __forceinline__ uint32_t lds_off(const void* p) {
  // Generic pointers to LDS carry the wave-relative LDS byte offset in the
  // low 32 bits (CDNA5 ISA 10.2).
  return (uint32_t)(uintptr_t)p;
}

__device__ __forceinline__ void async_store_from_lds_b32(void* gaddr, uint32_t loff) {
  // GLOBAL_STORE_ASYNC_FROM_LDS_B32: LDS -> global, tracked with ASYNCcnt.
  asm volatile("global_store_async_from_lds_b32 %0, %1, off"
               :: "v"((uint64_t)(uintptr_t)gaddr), "v"(loff)
               : "memory");
}

__device__ __forceinline__ void wave_argmax(float& m, int& i) {
#pragma unroll
  for (int off = 16; off > 0; off >>= 1) {
    float om = __shfl_xor(m, off, 32);
    int   oi = __shfl_xor(i, off, 32);
    if (om > m || (om == m && oi < i)) { m = om; i = oi; }
  }
}

template <bool FIRST>
__device__ __forceinline__ void sweep(const float* __restrict__ base,
                                      float* __restrict__ s_dist, int tid,
                                      float px, float py, float pz,
                                      float& bmax, int& bidx) {
  float m = -1.0f;
  int   mi = tid;
#pragma unroll 16
  for (int k = 0; k < FPS_PPT; ++k) {
    const int j = tid + k * FPS_THREADS;
    const F3 v = *reinterpret_cast<const F3*>(base + 3u * (unsigned)j);
    const float dx = v.x - px, dy = v.y - py, dz = v.z - pz;
    const float d  = dx * dx + dy * dy + dz * dz;   // lowers to v_pk_* math
    float nd;
    if (FIRST) {
      nd = d;                       // min(BIG, d) == d
    } else {
      nd = fminf(s_dist[j], d);     // ds_load_b32 (thread-private slot)
    }
    s_dist[j] = nd;                 // ds_store_b32
    if (nd > m) { m = nd; mi = j; } // strict > keeps lowest j (jnp.argmax tie rule)
  }
  bmax = m;
  bidx = mi;
}

__global__ __launch_bounds__(FPS_THREADS, 1)
void fps_kernel(const float* __restrict__ xyz,
                int* __restrict__ out_idx,
                float* __restrict__ out_pts) {
  const int b   = blockIdx.x;
  const int tid = threadIdx.x;
  const float* base = xyz + (size_t)b * FPS_N * 3;

  extern __shared__ float smem[];
  float* s_dist = smem;                         // FPS_N floats    (256 KB)
  float* s_px   = smem + FPS_N;                 // FPS_NPOINT      (8 KB)
  float* s_py   = s_px + FPS_NPOINT;            // FPS_NPOINT
  float* s_pz   = s_py + FPS_NPOINT;            // FPS_NPOINT
  int*   s_sidx = (int*)(s_pz + FPS_NPOINT);    // FPS_NPOINT ints

  __shared__ float s_rmax[32];
  __shared__ int   s_ridx[32];
  __shared__ int   s_pickv;

  __builtin_prefetch(base + 3 * tid, 0, 0);     // global_prefetch_b8: warm L2

  // Pick 0 is always index 0 (uniform address -> scalar loads).
  float px = base[0], py = base[1], pz = base[2];
  if (tid == 0) { s_sidx[0] = 0; s_px[0] = px; s_py[0] = py; s_pz[0] = pz; }

  const int wid  = tid >> 5;
  const int lane = tid & 31;

  for (int s = 1; s < FPS_NPOINT; ++s) {
    float bmax; int bidx;
    if (s == 1) sweep<true >(base, s_dist, tid, px, py, pz, bmax, bidx);
    else        sweep<false>(base, s_dist, tid, px, py, pz, bmax, bidx);

    // wave32 shuffle reduction, then 32-wave LDS reduction by wave 0.
    wave_argmax(bmax, bidx);
    if (lane == 0) { s_rmax[wid] = bmax; s_ridx[wid] = bidx; }
    __syncthreads();
    if (wid == 0) {
      float m = s_rmax[lane];
      int   i = s_ridx[lane];
      wave_argmax(m, i);
      if (lane == 0) s_pickv = i;
    }
    __syncthreads();

    const int pick = s_pickv;
    const F3 v = *reinterpret_cast<const F3*>(base + 3u * (unsigned)pick);
    px = v.x; py = v.y; pz = v.z;
    if (tid == 0) {
      s_sidx[s] = pick;
      s_px[s] = v.x; s_py[s] = v.y; s_pz[s] = v.z;
    }
  }

  __syncthreads();

  // Drain pick buffers LDS -> global with async stores (ASYNCcnt path).
  // d_out layout: [B*NPOINT int32 idx][B*3*NPOINT f32 sampled (x-row,y-row,z-row)]
  int*   oidx = out_idx + b * FPS_NPOINT;
  float* opts = out_pts + (size_t)b * 3 * FPS_NPOINT;
  for (int k = tid; k < FPS_NPOINT; k += FPS_THREADS) {
    async_store_from_lds_b32(oidx + k,                  lds_off(s_sidx + k));
    async_store_from_lds_b32(opts + k,                  lds_off(s_px + k));
    async_store_from_lds_b32(opts + FPS_NPOINT + k,     lds_off(s_py + k));
    async_store_from_lds_b32(opts + 2 * FPS_NPOINT + k, lds_off(s_pz + k));
  }
  asm volatile("s_wait_asynccnt 0x0" ::: "memory");
}

extern "C" void kernel_launch(void* const* d_in, const int* in_sizes, int n_in,
                              void* d_out, int out_size, void* d_ws, size_t ws_size,
                              hipStream_t stream) {
  const float* xyz = (const float*)d_in[0];
  const int B = in_sizes[0] / (FPS_N * 3);          // 16 for this harness

  int*   out_idx = (int*)d_out;                      // idx bits first
  float* out_pts = (float*)d_out + (size_t)B * FPS_NPOINT;

  const size_t smem = (size_t)(FPS_N + 4 * FPS_NPOINT) * sizeof(float); // 294912 B
  hipFuncSetAttribute(reinterpret_cast<const void*>(fps_kernel),
                      hipFuncAttributeMaxDynamicSharedMemorySize, (int)smem);

  fps_kernel<<<B, FPS_THREADS, smem, stream>>>(xyz, out_idx, out_pts);
}